// Low_Rank_HORNN_39290360824514
// MI455X (gfx1250) — compile-verified
//
#include <hip/hip_runtime.h>
#include <hip/hip_bf16.h>
#include <math.h>

typedef float v2f __attribute__((ext_vector_type(2)));
typedef float v8f __attribute__((ext_vector_type(8)));

#define B_ 128
#define T_ 1024
#define I_ 64
#define H_ 512
#define O_ 32
#define TAU_ 0.2f
#define NOISE_STD_ 0.05f

// ---------------------------------------------------------------------------
// Kernel A: input projection  inp[m,h] = u_flat[m,:] . Win_w[h,:] + Win_b[h]
// m = b*T + t (u is [B,T,I] row-major so u_flat[m*I + i]).
// WMMA f32 16x16x4: one wave owns a 16(M)x16(N) tile, K=64 in 16 steps of 4.
// A frags preloaded for the whole K so they are reused across all 32 N tiles.
// ---------------------------------------------------------------------------
__global__ __launch_bounds__(256) void k_inproj(const float* __restrict__ u,
                                                const float* __restrict__ Ww,
                                                const float* __restrict__ Wb,
                                                float* __restrict__ inp) {
  const int lane = threadIdx.x & 31;
  const int wave = threadIdx.x >> 5;
  const int l16  = lane & 15;
  const int hi   = lane >> 4;                 // 0: K {0,1}, 1: K {2,3}
  const int m0   = (blockIdx.x * 8 + wave) * 16;

  // A fragments: lane<16 -> A[M=l16][k, k+1], lane>=16 -> A[M=l16][k+2, k+3]
  v2f a[16];
#pragma unroll
  for (int kk = 0; kk < 16; ++kk) {
    const int k = kk * 4 + hi * 2;
    a[kk] = *(const v2f*)(u + (size_t)(m0 + l16) * I_ + k);
  }

#pragma unroll 1
  for (int nt = 0; nt < H_ / 16; ++nt) {
    const int n0 = nt * 16;
    const float bias = Wb[n0 + l16];
    v8f c = {bias, bias, bias, bias, bias, bias, bias, bias};
#pragma unroll
    for (int kk = 0; kk < 16; ++kk) {
      const int k = kk * 4 + hi * 2;
      // B[k][n] = Win_w[n][k]  (Win_w is [H,I] row-major)
      v2f b = *(const v2f*)(Ww + (size_t)(n0 + l16) * I_ + k);
      c = __builtin_amdgcn_wmma_f32_16x16x4_f32(false, a[kk], false, b,
                                                (short)0, c, false, false);
    }
#pragma unroll
    for (int j = 0; j < 8; ++j) {   // C vgpr j: M = j (lanes0-15) / j+8 (16-31)
      inp[(size_t)(m0 + j + hi * 8) * H_ + n0 + l16] = c[j];
    }
  }
}

// ---------------------------------------------------------------------------
// Kernel B: the recurrence. One 512-thread block per batch element; thread h
// owns hidden unit h; rank-2 weights live in registers. Per step: wave-level
// shfl_xor reductions, ONE barrier with phase double-buffered LDS partials,
// software-pipelined ip/noise loads, prefetch 4 steps ahead.
// ---------------------------------------------------------------------------
__global__ __launch_bounds__(512) void k_scan(const float* __restrict__ inp,
                                              const float* __restrict__ x0,
                                              const float* __restrict__ noise,
                                              const float* __restrict__ Mr,
                                              const float* __restrict__ Nr,
                                              const float* __restrict__ Ltb,
                                              const float* __restrict__ Mtb,
                                              const float* __restrict__ Ntb,
                                              float* __restrict__ traj,
                                              float* __restrict__ xfin) {
  const int b    = blockIdx.x;
  const int h    = threadIdx.x;
  const int lane = threadIdx.x & 31;
  const int wave = threadIdx.x >> 5;

  __shared__ float red[2][16][8];             // [phase][wave][6 partial sums]

  const float mr0 = Mr[h * 2 + 0], mr1 = Mr[h * 2 + 1];
  const float nr0 = Nr[h * 2 + 0], nr1 = Nr[h * 2 + 1];
  const float lt0 = Ltb[h * 2 + 0], lt1 = Ltb[h * 2 + 1];
  const float mt0 = Mtb[h * 2 + 0], mt1 = Mtb[h * 2 + 1];
  const float nt0 = Ntb[h * 2 + 0], nt1 = Ntb[h * 2 + 1];
  const float srn = 1.0f / (float)H_;
  const float stb = 1.0f / ((float)H_ * (float)H_);

  float x = x0[b * H_ + h];
  const float* ipp = inp   + (size_t)b * T_ * H_ + h;   // stride H_ per step
  const float* nzp = noise + (size_t)b * H_ + h;        // [T,B,H]: stride B_*H_
  float*       trp = traj  + (size_t)b * T_ * H_ + h;

  float ip = ipp[0];
  float nz = nzp[0];

  for (int t = 0; t < T_; ++t) {
    // pipeline next step's operands behind this step's reduction/barrier
    float ipn = 0.0f, nzn = 0.0f;
    if (t + 1 < T_) {
      ipn = ipp[(size_t)(t + 1) * H_];
      nzn = nzp[(size_t)(t + 1) * (B_ * H_)];
    }
    if (t + 4 < T_) {
      __builtin_prefetch(ipp + (size_t)(t + 4) * H_, 0, 0);
      __builtin_prefetch(nzp + (size_t)(t + 4) * (B_ * H_), 0, 0);
    }

    const float r = tanhf(x);
    float s0 = r * nr0, s1 = r * nr1;         // r @ N_rnn partials
    float s2 = r * mt0, s3 = r * mt1;         // r @ M_tb  partials
    float s4 = r * nt0, s5 = r * nt1;         // r @ N_tb  partials

#pragma unroll
    for (int off = 16; off > 0; off >>= 1) {
      s0 += __shfl_xor(s0, off, 32);
      s1 += __shfl_xor(s1, off, 32);
      s2 += __shfl_xor(s2, off, 32);
      s3 += __shfl_xor(s3, off, 32);
      s4 += __shfl_xor(s4, off, 32);
      s5 += __shfl_xor(s5, off, 32);
    }

    const int ph = t & 1;
    if (lane == 0) {
      red[ph][wave][0] = s0; red[ph][wave][1] = s1;
      red[ph][wave][2] = s2; red[ph][wave][3] = s3;
      red[ph][wave][4] = s4; red[ph][wave][5] = s5;
    }
    __syncthreads();   // single barrier/step; phase buffer prevents WAR races

    float a0 = 0.f, a1 = 0.f, p0 = 0.f, p1 = 0.f, q0 = 0.f, q1 = 0.f;
#pragma unroll
    for (int w = 0; w < 16; ++w) {            // broadcast reads (same addr/wave)
      a0 += red[ph][w][0]; a1 += red[ph][w][1];
      p0 += red[ph][w][2]; p1 += red[ph][w][3];
      q0 += red[ph][w][4]; q1 += red[ph][w][5];
    }

    const float rec = ip + srn * (a0 * mr0 + a1 * mr1)
                         + stb * ((p0 * q0) * lt0 + (p1 * q1) * lt1);
    x = x + NOISE_STD_ * nz + TAU_ * (rec - x);
    trp[(size_t)t * H_] = x;

    ip = ipn;
    nz = nzn;
  }
  xfin[b * H_ + h] = x;
}

// ---------------------------------------------------------------------------
// Kernel C: out[m,o] = tanh(traj[m,:]) . Wout_w[o,:] + Wout_b[o]
// WMMA f32 16x16x4, K=512; one wave owns 16 rows x both 16-wide N tiles
// (O=32) so each tanh'ed A fragment is reused twice.
// ---------------------------------------------------------------------------
__global__ __launch_bounds__(256) void k_outproj(const float* __restrict__ traj,
                                                 const float* __restrict__ Wo,
                                                 const float* __restrict__ Wob,
                                                 float* __restrict__ out) {
  const int lane = threadIdx.x & 31;
  const int wave = threadIdx.x >> 5;
  const int l16  = lane & 15;
  const int hi   = lane >> 4;
  const int m0   = (blockIdx.x * 8 + wave) * 16;

  const float b0v = Wob[l16];
  const float b1v = Wob[16 + l16];
  v8f c0 = {b0v, b0v, b0v, b0v, b0v, b0v, b0v, b0v};
  v8f c1 = {b1v, b1v, b1v, b1v, b1v, b1v, b1v, b1v};

#pragma unroll 4
  for (int kk = 0; kk < H_ / 4; ++kk) {
    const int k = kk * 4 + hi * 2;
    v2f a = *(const v2f*)(traj + (size_t)(m0 + l16) * H_ + k);
    a.x = tanhf(a.x);
    a.y = tanhf(a.y);
    v2f bb0 = *(const v2f*)(Wo + (size_t)(l16)      * H_ + k);
    v2f bb1 = *(const v2f*)(Wo + (size_t)(16 + l16) * H_ + k);
    c0 = __builtin_amdgcn_wmma_f32_16x16x4_f32(false, a, false, bb0,
                                               (short)0, c0, false, false);
    c1 = __builtin_amdgcn_wmma_f32_16x16x4_f32(false, a, false, bb1,
                                               (short)0, c1, false, false);
  }
#pragma unroll
  for (int j = 0; j < 8; ++j) {
    out[(size_t)(m0 + j + hi * 8) * O_ + l16]      = c0[j];
    out[(size_t)(m0 + j + hi * 8) * O_ + 16 + l16] = c1[j];
  }
}

// ---------------------------------------------------------------------------
// d_in order: u, x0, noise, Win_w, Win_b, Wout_w, Wout_b,
//             M_rnn, N_rnn, L_tb, M_tb, N_tb
// d_out: output [B,T,O] ++ x_final [B,H] ++ traj [B,T,H]   (fp32)
// d_ws: needs B*T*H*4 = 256 MiB for the precomputed input projection.
// ---------------------------------------------------------------------------
extern "C" void kernel_launch(void* const* d_in, const int* in_sizes, int n_in,
                              void* d_out, int out_size, void* d_ws, size_t ws_size,
                              hipStream_t stream) {
  const float* u      = (const float*)d_in[0];
  const float* x0     = (const float*)d_in[1];
  const float* noise  = (const float*)d_in[2];
  const float* Win_w  = (const float*)d_in[3];
  const float* Win_b  = (const float*)d_in[4];
  const float* Wout_w = (const float*)d_in[5];
  const float* Wout_b = (const float*)d_in[6];
  const float* M_rnn  = (const float*)d_in[7];
  const float* N_rnn  = (const float*)d_in[8];
  const float* L_tb   = (const float*)d_in[9];
  const float* M_tb   = (const float*)d_in[10];
  const float* N_tb   = (const float*)d_in[11];

  float* out  = (float*)d_out;
  float* xfin = out + (size_t)B_ * T_ * O_;
  float* traj = xfin + (size_t)B_ * H_;
  float* inp  = (float*)d_ws;                 // [B,T,H] fp32 scratch

  const int mtiles = (B_ * T_) / 16;          // 8192
  const int gemm_blocks = mtiles / 8;         // 8 waves per 256-thread block

  k_inproj<<<gemm_blocks, 256, 0, stream>>>(u, Win_w, Win_b, inp);
  k_scan<<<B_, 512, 0, stream>>>(inp, x0, noise, M_rnn, N_rnn, L_tb, M_tb,
                                 N_tb, traj, xfin);
  k_outproj<<<gemm_blocks, 256, 0, stream>>>(traj, Wout_w, Wout_b, out);
}